// PerlinNoise_43662637531391
// MI455X (gfx1250) — compile-verified
//
#include <hip/hip_runtime.h>
#include <cstdint>
#include <cstddef>

#define BLOCK 256

typedef unsigned int v4u __attribute__((ext_vector_type(4)));
typedef int          v4i __attribute__((ext_vector_type(4)));
typedef int          v8i __attribute__((ext_vector_type(8)));

__device__ __forceinline__ float wquint(float t) {
    // (10 + t*(6t - 15)) * t^3  -- reference's quintic smoothstep
    return (10.0f + t * (6.0f * t - 15.0f)) * (t * t * t);
}

// Dot the 8 gradient vectors (24 contiguous floats, 16B aligned) at one grid
// vertex against the offset vector (ox,oy,oz). Six b128 loads.
__device__ __forceinline__ void corner_dot(const float* __restrict__ v,
                                           float ox, float oy, float oz,
                                           float r[8]) {
    const float4* q = (const float4*)v;
    float4 q0 = q[0], q1 = q[1], q2 = q[2], q3 = q[3], q4 = q[4], q5 = q[5];
    float g[24] = {q0.x,q0.y,q0.z,q0.w, q1.x,q1.y,q1.z,q1.w,
                   q2.x,q2.y,q2.z,q2.w, q3.x,q3.y,q3.z,q3.w,
                   q4.x,q4.y,q4.z,q4.w, q5.x,q5.y,q5.z,q5.w};
    #pragma unroll
    for (int f = 0; f < 8; ++f)
        r[f] = fmaf(g[3*f+0], ox, fmaf(g[3*f+1], oy, g[3*f+2] * oz));
}

__global__ __launch_bounds__(BLOCK)
void PerlinNoise_kernel(const float* __restrict__ xin,
                        const float* __restrict__ grads,
                        float* __restrict__ out, int n) {
    __shared__ float sx[BLOCK * 3];

    const int  tid        = threadIdx.x;
    const long blockStart = (long)blockIdx.x * BLOCK;
    const long p          = blockStart + tid;

    // ---- Stage this block's 768 coordinates into LDS via the Tensor Data
    // Mover (wave 0 issues one DMA; TENSORcnt tracked, VMEM pipe stays free).
    if (tid < 32) {
        long remL = 3L * ((long)n - blockStart);
        int  rem  = (remL < 0) ? 0 : (remL > 3 * BLOCK ? 3 * BLOCK : (int)remL);

        uint64_t ga = (uint64_t)(uintptr_t)(xin + blockStart * 3);
        uint32_t la = (uint32_t)(uintptr_t)(&sx[0]);   // LDS byte offset

        // D# group 0: count=1 | lds_addr | global_addr[56:0] | type=2
        v4u g0 = { 1u,
                   la,
                   (uint32_t)ga,
                   (uint32_t)((ga >> 32) & 0x01FFFFFFu) | 0x80000000u };

        // D# group 1:
        //  w0: data_size=2 (4B) at [17:16]
        //  w1: tensor_dim0[15:0] at [31:16]  (= rem, OOB reads -> zero)
        //  w2: tensor_dim1[15:0]=1 at [31:16]
        //  w3: tile_dim0=768 at [31:16]
        //  w4: tile_dim1=1
        //  w5: tensor_dim0_stride=768
        //  w6: tensor_dim1_stride[15:0]=768 at [31:16]
        v8i g1 = { (int)0x00020000u,
                   (int)(((uint32_t)rem & 0xFFFFu) << 16),
                   (int)(1u   << 16),
                   (int)(768u << 16),
                   1,
                   768,
                   (int)(768u << 16),
                   0 };
        v4i gz = {0, 0, 0, 0};
#if defined(__clang_major__) && (__clang_major__ >= 23)
        v8i gz8 = {0, 0, 0, 0, 0, 0, 0, 0};
        __builtin_amdgcn_tensor_load_to_lds(g0, g1, gz, gz, gz8, 0);
#else
        __builtin_amdgcn_tensor_load_to_lds(g0, g1, gz, gz, 0);
#endif
        __builtin_amdgcn_s_wait_tensorcnt(0);
    }
    __syncthreads();

    if (p >= (long)n) return;

    // ---- Cell coordinates (match reference op-for-op: fmod, trunc).
    float px = fmodf(sx[tid * 3 + 0] * 64.0f, 64.0f);
    float py = fmodf(sx[tid * 3 + 1] * 64.0f, 64.0f);
    float pz = fmodf(sx[tid * 3 + 2] * 64.0f, 64.0f);
    float fx = truncf(px), fy = truncf(py), fz = truncf(pz);
    int   ix = (int)fx,    iy = (int)fy,    iz = (int)fz;
    float lx = px - fx,    ly = py - fy,    lz = pz - fz;
    float wx = wquint(lx), wy = wquint(ly), wz = wquint(lz);

    const int sZ = 8 * 3;          // 24 floats per vertex
    const int sY = 65 * sZ;
    const int sX = 65 * sY;
    const float* base = grads + (size_t)ix * sX + (size_t)iy * sY + (size_t)iz * sZ;

    // ---- 8 corner dots, then lerp x, then y, then z (reference order).
    float vzl[2][8];
    #pragma unroll
    for (int cz = 0; cz < 2; ++cz) {
        float oz = lz - (float)cz;
        float vxy[2][8];
        #pragma unroll
        for (int cy = 0; cy < 2; ++cy) {
            float oy = ly - (float)cy;
            const float* b0 = base + cy * sY + cz * sZ;
            float v0[8], v1[8];
            corner_dot(b0,      lx,        oy, oz, v0);
            corner_dot(b0 + sX, lx - 1.0f, oy, oz, v1);
            #pragma unroll
            for (int f = 0; f < 8; ++f)
                vxy[cy][f] = fmaf(wx, v1[f] - v0[f], v0[f]);
        }
        #pragma unroll
        for (int f = 0; f < 8; ++f)
            vzl[cz][f] = fmaf(wy, vxy[1][f] - vxy[0][f], vxy[0][f]);
    }
    float r[8];
    #pragma unroll
    for (int f = 0; f < 8; ++f)
        r[f] = fmaf(wz, vzl[1][f] - vzl[0][f], vzl[0][f]);

    float4* o = (float4*)(out + (size_t)p * 8);
    o[0] = make_float4(r[0], r[1], r[2], r[3]);
    o[1] = make_float4(r[4], r[5], r[6], r[7]);
}

extern "C" void kernel_launch(void* const* d_in, const int* in_sizes, int n_in,
                              void* d_out, int out_size, void* d_ws, size_t ws_size,
                              hipStream_t stream) {
    const float* x     = (const float*)d_in[0];   // (N, 3) f32
    const float* grads = (const float*)d_in[1];   // (65,65,65,8,3) f32
    float*       out   = (float*)d_out;           // (N, 8) f32

    int n      = in_sizes[0] / 3;
    int blocks = (n + BLOCK - 1) / BLOCK;
    PerlinNoise_kernel<<<blocks, BLOCK, 0, stream>>>(x, grads, out, n);
}